// QuestionAnsweringOutputLayer_73976516706781
// MI455X (gfx1250) — compile-verified
//
#include <hip/hip_runtime.h>
#include <hip/hip_bf16.h>
#include <math.h>

typedef float f4  __attribute__((ext_vector_type(4)));
typedef float v2f __attribute__((ext_vector_type(2)));
typedef float v8f __attribute__((ext_vector_type(8)));

#define NEG_INF (-__builtin_inff())

constexpr int Bb = 32, Ss = 2048, Dd = 1024;
constexpr int BS = Bb * Ss;

// ---------------------------------------------------------------------------
// Kernel 1: fused dual mat-vec.  Each wave handles 16 rows of the flattened
// [B*S, D] matrix; lane L covers row (L&15), D-half (L>>4).  Partial sums are
// reduced across the two half-lanes AND routed (start vs end) by a single
// V_WMMA_F32_16X16X4_F32 with a sparse selection A-matrix:
//   A[0,0]=A[0,2]=1  -> D[0,n] = acc_s(h0)+acc_s(h1)  (start dot, lanes 0-15)
//   A[1,1]=A[1,3]=1  -> D[1,n] = acc_e(h0)+acc_e(h1)  (end   dot, lanes 0-15)
// B VGPR0 = acc_s, VGPR1 = acc_e per the 4x16 B layout (K0/K1 lanes 0-15,
// K2/K3 lanes 16-31).
// ---------------------------------------------------------------------------
__global__ __launch_bounds__(256) void qa_scores_kernel(
    const float* __restrict__ ce, const int* __restrict__ attn,
    const float* __restrict__ Ws, const float* __restrict__ bs_p,
    const float* __restrict__ We, float* __restrict__ out)
{
    __shared__ float wlds[2 * Dd];           // [0:1024)=Ws, [1024:2048)=We[:D]
    const int t = threadIdx.x;

    // 256 threads x one float4 each fills each 1024-float weight vector
    ((f4*)wlds)[t]       = ((const f4*)Ws)[t];
    ((f4*)wlds)[256 + t] = ((const f4*)We)[t];
    __syncthreads();

    const int lane    = t & 31;
    const int wave    = (blockIdx.x * 256 + t) >> 5;   // global wave id
    const int rowBase = wave * 16;
    const int row     = rowBase + (lane & 15);
    const int half    = lane >> 4;                     // 0 or 1

    const f4* __restrict__ cp  = (const f4*)(ce + (size_t)row * Dd + half * 512);
    const f4* __restrict__ wsp = (const f4*)(wlds + half * 512);
    const f4* __restrict__ wep = (const f4*)(wlds + Dd + half * 512);

    float acc_s = 0.0f, acc_e = 0.0f;
#pragma unroll 8
    for (int i = 0; i < 128; ++i) {
        f4 c = __builtin_nontemporal_load(cp + i);  // 256MB stream: bypass-ish L2
        f4 w = wsp[i];
        f4 v = wep[i];
        acc_s += c.x * w.x + c.y * w.y + c.z * w.z + c.w * w.w;
        acc_e += c.x * v.x + c.y * v.y + c.z * v.z + c.w * v.w;
    }

    // --- WMMA cross-lane reduction (EXEC all ones here) ---
    v2f A;
    A.x = (lane == 0 || lane == 16) ? 1.0f : 0.0f;   // A[0,0], A[0,2]
    A.y = (lane == 1 || lane == 17) ? 1.0f : 0.0f;   // A[1,1], A[1,3]
    v2f Bm;
    Bm.x = acc_s;                                    // B[K even rows, n]
    Bm.y = acc_e;                                    // B[K odd  rows, n]
    v8f C = {};
    v8f Dm = __builtin_amdgcn_wmma_f32_16x16x4_f32(
        false, A, false, Bm, (short)0, C, false, false);

    if (lane < 16) {
        const int r   = rowBase + lane;
        const float b = bs_p[0];
        const int  m  = attn[r];
        out[r]      = m ? (Dm[0] + b) : NEG_INF;     // start_scores (masked)
        out[BS + r] = Dm[1];                         // raw ce.We[:D] partial
    }
}

// ---------------------------------------------------------------------------
// Kernel 2: one block per batch.  max/argmax -> start softmax -> best-start
// dot -> end scores (in-place over the partial) -> end softmax.
// ---------------------------------------------------------------------------
__global__ __launch_bounds__(256) void qa_softmax_kernel(
    const float* __restrict__ ce, const int* __restrict__ attn,
    const float* __restrict__ We, const float* __restrict__ be_p,
    float* __restrict__ out)
{
    __shared__ float rv[256];
    __shared__ int   ri[256];
    const int b = blockIdx.x, t = threadIdx.x;
    const float* ss = out + (size_t)b * Ss;          // masked start scores

    // ---- max + first-occurrence argmax over S ----
    float m = NEG_INF; int mi = 0;
    for (int s = t; s < Ss; s += 256) {
        float v = ss[s];
        if (v > m) { m = v; mi = s; }
    }
    rv[t] = m; ri[t] = mi; __syncthreads();
    for (int o = 128; o; o >>= 1) {
        if (t < o) {
            float v = rv[t + o]; int i = ri[t + o];
            if (v > rv[t] || (v == rv[t] && i < ri[t])) { rv[t] = v; ri[t] = i; }
        }
        __syncthreads();
    }
    const float smax = rv[0];
    const int   best = ri[0];
    __syncthreads();

    // ---- start softmax ----
    float p = 0.0f;
    for (int s = t; s < Ss; s += 256) p += expf(ss[s] - smax);
    rv[t] = p; __syncthreads();
    for (int o = 128; o; o >>= 1) { if (t < o) rv[t] += rv[t + o]; __syncthreads(); }
    const float sinv = 1.0f / rv[0];
    __syncthreads();
    for (int s = t; s < Ss; s += 256)
        out[2 * BS + (size_t)b * Ss + s] = expf(ss[s] - smax) * sinv;

    // ---- dot(ce[b,best,:], We[D:2D]) : 256 threads x one float4 ----
    const f4* cb = (const f4*)(ce + ((size_t)b * Ss + best) * Dd);
    const f4* w2 = (const f4*)(We + Dd);
    f4 c = cb[t];
    f4 w = w2[t];
    rv[t] = c.x * w.x + c.y * w.y + c.z * w.z + c.w * w.w;
    __syncthreads();
    for (int o = 128; o; o >>= 1) { if (t < o) rv[t] += rv[t + o]; __syncthreads(); }
    const float bestdot = rv[0];
    __syncthreads();

    // ---- finalize end scores in-place, track max ----
    const float be = be_p[0];
    float* es = out + BS + (size_t)b * Ss;           // holds raw partial now
    float em = NEG_INF;
    for (int s = t; s < Ss; s += 256) {
        const bool valid = (attn[(size_t)b * Ss + s] != 0) && !(best == 0 && s == 0);
        const float e = valid ? (es[s] + bestdot + be) : NEG_INF;
        es[s] = e;                                   // same thread re-reads below
        if (e > em) em = e;
    }
    rv[t] = em; __syncthreads();
    for (int o = 128; o; o >>= 1) {
        if (t < o) { if (rv[t + o] > rv[t]) rv[t] = rv[t + o]; }
        __syncthreads();
    }
    const float emax = rv[0];
    __syncthreads();

    float q = 0.0f;
    for (int s = t; s < Ss; s += 256) q += expf(es[s] - emax);
    rv[t] = q; __syncthreads();
    for (int o = 128; o; o >>= 1) { if (t < o) rv[t] += rv[t + o]; __syncthreads(); }
    const float einv = 1.0f / rv[0];
    for (int s = t; s < Ss; s += 256)
        out[3 * BS + (size_t)b * Ss + s] = expf(es[s] - emax) * einv;
}

extern "C" void kernel_launch(void* const* d_in, const int* in_sizes, int n_in,
                              void* d_out, int out_size, void* d_ws, size_t ws_size,
                              hipStream_t stream) {
    (void)in_sizes; (void)n_in; (void)out_size; (void)d_ws; (void)ws_size;
    const float* ce  = (const float*)d_in[0];   // [B,S,D] f32
    const int*   am  = (const int*)  d_in[1];   // [B,S]   bool->int
    const float* Ws  = (const float*)d_in[2];   // [D]
    const float* bs  = (const float*)d_in[3];   // [1]
    const float* We  = (const float*)d_in[4];   // [2D]
    const float* be  = (const float*)d_in[5];   // [1]
    float* out = (float*)d_out;                 // 4 x [B,S] concatenated

    // 65536 rows, 16 rows/wave, 8 waves/block -> 512 blocks
    qa_scores_kernel<<<BS / 128, 256, 0, stream>>>(ce, am, Ws, bs, We, out);
    qa_softmax_kernel<<<Bb, 256, 0, stream>>>(ce, am, We, be, out);
}